// MIRT_455266533950
// MI455X (gfx1250) — compile-verified
//
#include <hip/hip_runtime.h>
#include <math.h>

typedef __attribute__((ext_vector_type(2))) float v2f;
typedef __attribute__((ext_vector_type(8))) float v8f;

#define BDIM 256
#define WAVES_PER_BLOCK (BDIM / 32)          // 8 waves (wave32)
#define PAIRS_PER_WAVE 16
#define PAIRS_PER_BLOCK (WAVES_PER_BLOCK * PAIRS_PER_WAVE)  // 128
#define KDIM 64
#define NB 1048576
#define NBLOCKS (NB / PAIRS_PER_BLOCK)       // 8192

// Kernel 1: each wave handles 16 (u,i,s) pairs. The 16 per-pair K=64 dot
// products are computed as the diagonal of a 16x16 f32 WMMA accumulator via
// 16 chained V_WMMA_F32_16X16X4_F32 ops (A = disc rows, B = theta rows^T).
// Then stable BCE-with-logits per pair, fixed-order reduction to one partial
// sum per block.
__global__ __launch_bounds__(BDIM) void mirt_wmma_kernel(
    const int* __restrict__ u_idx, const int* __restrict__ i_idx,
    const int* __restrict__ s_lab, const float* __restrict__ diff,
    const float* __restrict__ disc, const float* __restrict__ theta,
    float* __restrict__ partial)
{
    const int tid  = threadIdx.x;
    const int lane = tid & 31;
    const int wid  = tid >> 5;
    const int sub  = lane & 15;   // pair slot within the wave's 16-pair tile
    const int hh   = lane >> 4;   // half-wave id -> K sub-column (0 or 2)

    const int p = (blockIdx.x * WAVES_PER_BLOCK + wid) * PAIRS_PER_WAVE + sub;

    const int item = i_idx[p];
    const int user = u_idx[p];

    // Per-lane base: row start + 2*hh columns, in float2 (8-byte) units.
    const v2f* __restrict__ dp = (const v2f*)disc  + ((size_t)item * (KDIM / 2) + hh);
    const v2f* __restrict__ tp = (const v2f*)theta + ((size_t)user * (KDIM / 2) + hh);

    v8f c = {};
#pragma unroll
    for (int j = 0; j < KDIM / 4; ++j) {
        // A(16x4): lane holds disc[item][4j + 2*hh .. +1]
        v2f a = dp[2 * j];
        // B(4x16) = theta^T: identical per-lane addressing into theta[user]
        v2f b = tp[2 * j];
        // 8 args: (neg_a, A, neg_b, B, c_mod, C, reuse_a, reuse_b)
        c = __builtin_amdgcn_wmma_f32_16x16x4_f32(
                false, a, false, b, (short)0, c, false, false);
    }

    // Diagonal D[m][m]: lanes 0..7 -> c[lane] (m = lane),
    //                   lanes 24..31 -> c[lane-24] (m = lane-16 = sub).
    const bool valid = (lane < 8) || (lane >= 24);
    const int  r     = (lane < 8) ? lane : (lane - 24);
    float x = c[0];
#pragma unroll
    for (int q = 1; q < 8; ++q) x = (r == q) ? c[q] : x;

    x += diff[item];

    // loss = -(real*log_sigmoid(x) + (1-real)*log_sigmoid(-x))
    // log_sigmoid(x) = min(x,0) - log1p(exp(-|x|));  ls(-x) = ls(x) - x
    // => contrib = (1-real)*x - ls(x)
    const float real = (float)s_lab[p];
    const float t    = log1pf(expf(-fabsf(x)));
    const float ls   = fminf(x, 0.0f) - t;
    float contrib    = (1.0f - real) * x - ls;
    contrib = valid ? contrib : 0.0f;

    // Fixed-order wave reduction (wave32)
#pragma unroll
    for (int off = 16; off >= 1; off >>= 1)
        contrib += __shfl_xor(contrib, off, 32);

    __shared__ float wsum[WAVES_PER_BLOCK];
    if (lane == 0) wsum[wid] = contrib;
    __syncthreads();
    if (tid == 0) {
        float bsum = 0.0f;
#pragma unroll
        for (int w = 0; w < WAVES_PER_BLOCK; ++w) bsum += wsum[w];
        partial[blockIdx.x] = bsum;
    }
}

// Kernel 2: deterministic single-block reduction of the 8192 partials.
__global__ __launch_bounds__(256) void mirt_reduce_kernel(
    const float* __restrict__ partial, float* __restrict__ out,
    int n, float scale)
{
    __shared__ float sm[256];
    float s = 0.0f;
    for (int idx = threadIdx.x; idx < n; idx += 256) s += partial[idx];
    sm[threadIdx.x] = s;
    __syncthreads();
#pragma unroll
    for (int off = 128; off > 0; off >>= 1) {
        if ((int)threadIdx.x < off) sm[threadIdx.x] += sm[threadIdx.x + off];
        __syncthreads();
    }
    if (threadIdx.x == 0) out[0] = sm[0] * scale;
}

extern "C" void kernel_launch(void* const* d_in, const int* in_sizes, int n_in,
                              void* d_out, int out_size, void* d_ws, size_t ws_size,
                              hipStream_t stream) {
    (void)in_sizes; (void)n_in; (void)out_size; (void)ws_size;
    const int*   u     = (const int*)d_in[0];
    const int*   i     = (const int*)d_in[1];
    const int*   s     = (const int*)d_in[2];
    const float* diff  = (const float*)d_in[3];
    const float* disc  = (const float*)d_in[4];
    const float* theta = (const float*)d_in[5];
    float* partial = (float*)d_ws;   // needs NBLOCKS*4 = 32 KB of scratch

    mirt_wmma_kernel<<<NBLOCKS, BDIM, 0, stream>>>(u, i, s, diff, disc, theta, partial);
    mirt_reduce_kernel<<<1, 256, 0, stream>>>(partial, (float*)d_out,
                                              NBLOCKS, 1.0f / (float)NB);
}